// AdaptiveConv_79972291051911
// MI455X (gfx1250) — compile-verified
//
#include <hip/hip_runtime.h>

// Problem constants (from reference): B=8, C=64, H=W=256, K=5, x is (B,C,260,260)
#define NB   8
#define NC   64
#define NH   256
#define NW   256
#define XWD  260   // padded width/height of x

// 4 output rows per block; need 8 x-rows staged (h0 .. h0+7)
#define TH     4
#define XROWS  (TH + 4)
#define SEGS   (XWD / 4)            // 65 float4 segments per x row
#define CHUNKS (XROWS * SEGS)       // 520 float4 chunks per channel tile

#ifndef __has_builtin
#define __has_builtin(x) 0
#endif

#if __has_builtin(__builtin_amdgcn_global_load_async_to_lds_b128)
#define HAVE_ASYNC_LDS 1
typedef int v4i __attribute__((vector_size(4 * sizeof(int))));
typedef __attribute__((address_space(1))) v4i as1_v4i_t;   // global
typedef __attribute__((address_space(3))) v4i as3_v4i_t;   // LDS
#else
#define HAVE_ASYNC_LDS 0
#endif

__device__ __forceinline__ void wait_async_then_barrier() {
#if HAVE_ASYNC_LDS
#if __has_builtin(__builtin_amdgcn_s_wait_asynccnt)
    __builtin_amdgcn_s_wait_asynccnt(0);
#else
    asm volatile("s_wait_asynccnt 0" ::: "memory");
#endif
#endif
    __syncthreads();
}

__global__ __launch_bounds__(256)
void AdaptiveConv_gfx1250_kernel(const float* __restrict__ x,
                                 const float* __restrict__ ker,
                                 float* __restrict__ out)
{
    // Double-buffered x tile: 2 * 8 rows * 260 floats = 16,640 B of LDS
    __shared__ float lds[2][XROWS * XWD];

    const int tid = threadIdx.x;
    const int wq  = tid & 63;          // which group of 4 w's
    const int hr  = tid >> 6;          // 0..3 row inside tile
    const int w0  = wq * 4;

    const int blk = blockIdx.x;        // 0 .. 8*64-1
    const int b   = blk >> 6;
    const int h0  = (blk & 63) * TH;
    const int h   = h0 + hr;

    // ---- Load 25 per-pixel taps for this thread's 4 pixels; reused for all 64 channels ----
    float4 kv[25];
    {
        const float* kb = ker + (((size_t)b * 25) * NH + h) * NW + w0;
        #pragma unroll
        for (int idx = 0; idx < 25; ++idx) {
            kv[idx] = *(const float4*)(kb + (size_t)idx * NH * NW);
        }
    }

    const float* xbase = x + ((size_t)b * NC) * (size_t)XWD * XWD;

    // ---- Stage one channel's 8x260 tile into LDS buffer `buf` ----
    auto stage = [&](int c, int buf) {
        const float* xc = xbase + (size_t)c * XWD * XWD + (size_t)h0 * XWD;
        for (int t = tid; t < CHUNKS; t += 256) {
            const int row = t / SEGS;
            const int seg = t - row * SEGS;
            const float* gp = xc + row * XWD + seg * 4;
            float*       lp = &lds[buf][row * XWD + seg * 4];
#if HAVE_ASYNC_LDS
            __builtin_amdgcn_global_load_async_to_lds_b128(
                (as1_v4i_t*)(void*)gp, (as3_v4i_t*)(void*)lp, 0, 0);
#else
            *(float4*)lp = *(const float4*)gp;
#endif
        }
    };

    // Prologue: stage channel 0
    stage(0, 0);
    wait_async_then_barrier();

    for (int c = 0; c < NC; ++c) {
        const int cur = c & 1;
        if (c + 1 < NC) stage(c + 1, cur ^ 1);   // overlap next channel's copy

        // ---- Compute 4 outputs from the staged tile ----
        float a0 = 0.f, a1 = 0.f, a2 = 0.f, a3 = 0.f;
        #pragma unroll
        for (int i = 0; i < 5; ++i) {
            const float* lrow = &lds[cur][(hr + i) * XWD + w0];
            const float4 xa = *(const float4*)lrow;        // w0..w0+3
            const float4 xb = *(const float4*)(lrow + 4);  // w0+4..w0+7
            const float win[8] = {xa.x, xa.y, xa.z, xa.w, xb.x, xb.y, xb.z, xb.w};
            #pragma unroll
            for (int j = 0; j < 5; ++j) {
                const float4 kk = kv[i * 5 + j];
                a0 = fmaf(win[j + 0], kk.x, a0);
                a1 = fmaf(win[j + 1], kk.y, a1);
                a2 = fmaf(win[j + 2], kk.z, a2);
                a3 = fmaf(win[j + 3], kk.w, a3);
            }
        }

        float* op = out + (((size_t)(b * NC + c) * NH + h) * NW + w0);
        *(float4*)op = make_float4(a0, a1, a2, a3);

        // Ensure next channel's tile landed and everyone finished reading `cur`
        wait_async_then_barrier();
    }
}

extern "C" void kernel_launch(void* const* d_in, const int* in_sizes, int n_in,
                              void* d_out, int out_size, void* d_ws, size_t ws_size,
                              hipStream_t stream)
{
    const float* x   = (const float*)d_in[0];
    const float* ker = (const float*)d_in[1];
    float*       out = (float*)d_out;

    dim3 grid(NB * (NH / TH));   // 512 blocks
    dim3 block(256);
    hipLaunchKernelGGL(AdaptiveConv_gfx1250_kernel, grid, block, 0, stream,
                       x, ker, out);
}